// ContextBasedLinear_2929167695912
// MI455X (gfx1250) — compile-verified
//
#include <hip/hip_runtime.h>

typedef float v2f __attribute__((ext_vector_type(2)));
typedef float v4f __attribute__((ext_vector_type(4)));
typedef float v8f __attribute__((ext_vector_type(8)));

#define B_DIM 64
#define N_DIM 1024
#define D_DIM 512

// ---------------------------------------------------------------------------
// Stage 1: partial column sums via V_WMMA_F32_16X16X4_F32 with A = ones.
//   D(16x16) = ones(16x4) x B(4x16) + C  ==> every row of D is colsum(B).
//   Row->K-slot permutation inside B cannot change a column sum, so the
//   mapping is robust; only N = lane%16 matters (documented layout).
// grid = (64 batches, NS row-splits, 4), block = 256 threads = 8 waves.
// Wave w of block z covers the 16 columns of tile t = z*8 + w (32 tiles).
// ---------------------------------------------------------------------------
__global__ void __launch_bounds__(256)
cbl_colsum_wmma(const float* __restrict__ x, float* __restrict__ part,
                int rowsPerSplit)
{
    const int lane = threadIdx.x & 31;
    const int wave = threadIdx.x >> 5;
    const int b    = blockIdx.x;
    const int ns   = blockIdx.y;
    const int tile = blockIdx.z * 8 + wave;      // 0..31, 16 cols each
    const int col  = (tile << 4) | (lane & 15);  // N = lane % 16
    const int half = lane >> 4;                  // rows {0,1} vs {2,3} of each group

    const float* p = x + ((size_t)b * N_DIM + (size_t)ns * rowsPerSplit
                          + 2 * half) * D_DIM + col;

    v2f a;  a[0] = 1.0f; a[1] = 1.0f;            // A-matrix: all ones
    v8f acc = {0.f, 0.f, 0.f, 0.f, 0.f, 0.f, 0.f, 0.f};

    for (int r = 0; r < rowsPerSplit; r += 4) {
        v2f bm;
        bm[0] = p[0];          // row g + 2*half
        bm[1] = p[D_DIM];      // row g + 2*half + 1
#if defined(__gfx1250__)
        // 8 args: (neg_a, A, neg_b, B, c_mod, C, reuse_a, reuse_b)
        acc = __builtin_amdgcn_wmma_f32_16x16x4_f32(
            false, a, false, bm, (short)0, acc, false, false);
#else
        acc[0] += bm[0] + bm[1] + (a[0] - 1.0f);   // host-pass placeholder
#endif
        p += 4 * D_DIM;
    }

    // All rows of D are identical; VGPR0 lane n (n<16) holds colsum for N=n.
    if (lane < 16)
        part[(((size_t)b * gridDim.y + ns) << 9) + (tile << 4) + lane] = acc[0];
}

// ---------------------------------------------------------------------------
// Stage 2: fold NS split-partials and pre-scale by gamma.  32768 threads.
// ---------------------------------------------------------------------------
__global__ void __launch_bounds__(256)
cbl_combine(const float* __restrict__ part, const float* __restrict__ gamma,
            float* __restrict__ sgam, int NS)
{
    const int i = blockIdx.x * blockDim.x + threadIdx.x;   // [0, 64*512)
    if (i < B_DIM * D_DIM) {
        const int b = i >> 9;
        const int d = i & 511;
        float s = 0.0f;
        for (int ns = 0; ns < NS; ++ns)
            s += part[(((size_t)b * NS + ns) << 9) + d];
        sgam[i] = s * gamma[0];
    }
}

// ---------------------------------------------------------------------------
// Stage 3: out = mu*x + sgam[b,d], float4-vectorized (clang ext_vector so the
// non-temporal store builtin accepts it).  x is L2-resident from stage 1;
// non-temporal stores keep the output stream from evicting it.
// ---------------------------------------------------------------------------
__global__ void __launch_bounds__(256)
cbl_axpy(const v4f* __restrict__ x4, const float* __restrict__ mu,
         const float* __restrict__ sgam, v4f* __restrict__ out4)
{
    const size_t i    = (size_t)blockIdx.x * blockDim.x + threadIdx.x;
    const size_t flat = i << 2;                     // element index
    const int d = (int)(flat & 511);
    const int b = (int)(flat >> 19);                // 1024*512 = 2^19
    const float m  = mu[0];
    const v4f   sg = *(const v4f*)(sgam + ((size_t)b << 9) + d);
    const v4f   xv = x4[i];
    v4f o;
    o[0] = fmaf(m, xv[0], sg[0]);
    o[1] = fmaf(m, xv[1], sg[1]);
    o[2] = fmaf(m, xv[2], sg[2]);
    o[3] = fmaf(m, xv[3], sg[3]);
    __builtin_nontemporal_store(o, &out4[i]);
}

// ---------------------------------------------------------------------------
// Fallback (only if ws_size is tiny): each thread owns one (b,d) column,
// sums 1024 rows, then rewrites its column.  grid = (64, 4), block = 128.
// ---------------------------------------------------------------------------
__global__ void __launch_bounds__(128)
cbl_fused_fallback(const float* __restrict__ x, const float* __restrict__ mu,
                   const float* __restrict__ gamma, float* __restrict__ out)
{
    const int b = blockIdx.x;
    const int d = blockIdx.y * 128 + threadIdx.x;
    const float* col = x + (size_t)b * N_DIM * D_DIM + d;
    float s = 0.0f;
    for (int n = 0; n < N_DIM; ++n) s += col[(size_t)n * D_DIM];
    const float sg = s * gamma[0];
    const float m  = mu[0];
    float* ocol = out + (size_t)b * N_DIM * D_DIM + d;
    for (int n = 0; n < N_DIM; ++n)
        ocol[(size_t)n * D_DIM] = fmaf(m, col[(size_t)n * D_DIM], sg);
}

extern "C" void kernel_launch(void* const* d_in, const int* in_sizes, int n_in,
                              void* d_out, int out_size, void* d_ws, size_t ws_size,
                              hipStream_t stream)
{
    const float* x     = (const float*)d_in[0];
    const float* mu    = (const float*)d_in[1];
    const float* gamma = (const float*)d_in[2];
    float*       out   = (float*)d_out;
    float*       ws    = (float*)d_ws;

    // Pick the number of n-splits that fits in scratch (needs partials + sgam).
    int NS = 8;
    while (NS > 1 &&
           (size_t)(B_DIM * NS * D_DIM + B_DIM * D_DIM) * sizeof(float) > ws_size)
        NS >>= 1;

    if ((size_t)(B_DIM * NS * D_DIM + B_DIM * D_DIM) * sizeof(float) > ws_size) {
        // Scratch-free fallback (still deterministic).
        cbl_fused_fallback<<<dim3(B_DIM, 4), 128, 0, stream>>>(x, mu, gamma, out);
        return;
    }

    float* part = ws;
    float* sgam = ws + (size_t)B_DIM * NS * D_DIM;

    dim3 g1(B_DIM, NS, 4);
    cbl_colsum_wmma<<<g1, 256, 0, stream>>>(x, part, N_DIM / NS);

    cbl_combine<<<(B_DIM * D_DIM + 255) / 256, 256, 0, stream>>>(part, gamma, sgam, NS);

    const size_t nvec4 = (size_t)B_DIM * N_DIM * D_DIM / 4;   // 8,388,608
    cbl_axpy<<<(unsigned)(nvec4 / 256), 256, 0, stream>>>(
        (const v4f*)x, mu, sgam, (v4f*)out);
}